// CPNet_33165737460025
// MI455X (gfx1250) — compile-verified
//
#include <hip/hip_runtime.h>
#include <hip/hip_bf16.h>

#define Bb 8
#define Nn 4096

typedef float v2f __attribute__((ext_vector_type(2)));
typedef float v8f __attribute__((ext_vector_type(8)));

// ---------------------------------------------------------------------------
// Kernel 1: fused dot4-WMMA + online softmax + weighted gather.
//   One wave (32 lanes) owns a 16-row tile of pc1; loops over all 256
//   16-column tiles of pc2. V_WMMA_F32_16X16X4_F32 computes all 256 dot4
//   products of a tile pair in one instruction.
//   A frag (16x4 f32): lanes 0-15 hold K=0,1 ; lanes 16-31 hold K=2,3 (M=lane&15)
//   B frag (4x16 f32): lanes 0-15 hold K=0,1 ; lanes 16-31 hold K=2,3 (N=lane&15)
//   C/D (16x16 f32):   VGPR r -> row r (lanes 0-15) / row r+8 (lanes 16-31),
//                      lane&15 -> column.
//   Each lane runs a PRIVATE online softmax over its own column subset
//   (no shuffles in the hot loop); the 16 per-lane states are merged at the
//   end with the standard flash-attention state-combine.
// ---------------------------------------------------------------------------
__global__ __launch_bounds__(128) void k1_flash(const float* __restrict__ pc1,
                                                const float* __restrict__ pc2,
                                                float* __restrict__ pcn,
                                                float* __restrict__ dist) {
  const int wave = threadIdx.x >> 5;
  const int lane = threadIdx.x & 31;
  const int half = lane >> 4;      // 0: lanes 0-15, 1: lanes 16-31
  const int lr   = lane & 15;
  const int bpS  = Nn / 64;        // row-blocks per sample (4 waves x 16 rows)
  const int b    = blockIdx.x / bpS;
  const int r0   = (blockIdx.x % bpS) * 64 + wave * 16;

  const float* __restrict__ P1 = pc1 + (size_t)b * 4 * Nn;
  const float* __restrict__ P2 = pc2 + (size_t)b * 4 * Nn;

  // A fragment: row M = r0+lr, K pair (0,1) or (2,3). Row 3 of pc1 is 1.0.
  v2f afrag;
  afrag.x = P1[(2 * half + 0) * Nn + r0 + lr];
  afrag.y = P1[(2 * half + 1) * Nn + r0 + lr];

  // Rows this lane owns in the C/D layout.
  float p1x[8], p1y[8], p1z[8], n1v[8];
  float m[8], l[8], ax[8], ay[8], az[8];
#pragma unroll
  for (int r = 0; r < 8; ++r) {
    const int row = r0 + 8 * half + r;
    p1x[r] = P1[0 * Nn + row];
    p1y[r] = P1[1 * Nn + row];
    p1z[r] = P1[2 * Nn + row];
    n1v[r] = p1x[r] * p1x[r] + p1y[r] * p1y[r] + p1z[r] * p1z[r];
    m[r] = -__builtin_inff();
    l[r] = 0.f; ax[r] = 0.f; ay[r] = 0.f; az[r] = 0.f;
  }

  for (int c0 = 0; c0 < Nn; c0 += 16) {
    if (c0 + 16 < Nn) {  // hint next tile into cache (global_prefetch_b8)
      __builtin_prefetch(&P2[0 * Nn + c0 + 16 + lr], 0, 3);
    }
    const float bx = P2[0 * Nn + c0 + lr];
    const float by = P2[1 * Nn + c0 + lr];
    const float bz = P2[2 * Nn + c0 + lr];
    v2f bfrag;
    bfrag.x = half ? bz : bx;
    bfrag.y = half ? 1.0f : by;    // pc2 row 3 == 1.0
    const float n2v = bx * bx + by * by + bz * bz;

    v8f D = {0.f, 0.f, 0.f, 0.f, 0.f, 0.f, 0.f, 0.f};
    // D[i][j] = dot4(p1_i, p2_j) = dot3 + 1
    D = __builtin_amdgcn_wmma_f32_16x16x4_f32(false, afrag, false, bfrag,
                                              (short)0, D, false, false);
#pragma unroll
    for (int r = 0; r < 8; ++r) {
      float d = n1v[r] + n2v - 2.0f * (D[r] - 1.0f);
      d = fmaxf(d, 1e-5f);                          // clip(d, EPS)
      const float s = 2.0f * __builtin_amdgcn_rcpf(d);  // FACT/d, v_rcp_f32
      const float mnew  = fmaxf(m[r], s);
      const float scale = __expf(m[r] - mnew);      // 1.0 when max unchanged
      const float w     = __expf(s - mnew);
      l[r]  = l[r]  * scale + w;
      ax[r] = ax[r] * scale + w * bx;
      ay[r] = ay[r] * scale + w * by;
      az[r] = az[r] * scale + w * bz;
      m[r] = mnew;
    }
  }

  // Merge the 16 per-lane online-softmax states within each half; then
  // lane 0 / lane 16 write their 8 rows.
#pragma unroll
  for (int r = 0; r < 8; ++r) {
    float mm = m[r], ls = l[r], xs = ax[r], ys = ay[r], zs = az[r];
    for (int off = 1; off <= 8; off <<= 1) {
      const float mo = __shfl_xor(mm, off, 32);
      const float lo = __shfl_xor(ls, off, 32);
      const float xo = __shfl_xor(xs, off, 32);
      const float yo = __shfl_xor(ys, off, 32);
      const float zo = __shfl_xor(zs, off, 32);
      const float mn = fmaxf(mm, mo);
      const float ea = __expf(mm - mn);
      const float eb = __expf(mo - mn);
      ls = ls * ea + lo * eb;
      xs = xs * ea + xo * eb;
      ys = ys * ea + yo * eb;
      zs = zs * ea + zo * eb;
      mm = mn;
    }
    const float inv = 1.0f / ls;
    const float nx = xs * inv, ny = ys * inv, nz = zs * inv;
    if (lr == 0) {
      const int row = r0 + 8 * half + r;
      pcn[((size_t)b * 3 + 0) * Nn + row] = nx;
      pcn[((size_t)b * 3 + 1) * Nn + row] = ny;
      pcn[((size_t)b * 3 + 2) * Nn + row] = nz;
      const float dx = p1x[r] - nx, dy = p1y[r] - ny, dz = p1z[r] - nz;
      dist[(size_t)b * Nn + row] = sqrtf(dx * dx + dy * dy + dz * dz);
    }
  }
}

// ---------------------------------------------------------------------------
// Kernel 2: per-sample mean/std (ddof=1) of pc1 and pc_nearest + mean(dist).
// stats[b*16 + 0..12] = m1xyz, s1xyz, m2xyz, s2xyz, dist_mean
// ---------------------------------------------------------------------------
__global__ __launch_bounds__(256) void k2_stats(const float* __restrict__ pc1,
                                                const float* __restrict__ pcn,
                                                const float* __restrict__ dist,
                                                float* __restrict__ stats) {
  const int b = blockIdx.x, tid = threadIdx.x;
  const float* P1 = pc1  + (size_t)b * 4 * Nn;
  const float* PN = pcn  + (size_t)b * 3 * Nn;
  const float* DI = dist + (size_t)b * Nn;

  float acc[13];
#pragma unroll
  for (int k = 0; k < 13; ++k) acc[k] = 0.f;
  for (int n = tid; n < Nn; n += 256) {
    const float x1 = P1[n], y1 = P1[Nn + n], z1 = P1[2 * Nn + n];
    const float x2 = PN[n], y2 = PN[Nn + n], z2 = PN[2 * Nn + n];
    acc[0] += x1; acc[1] += y1; acc[2] += z1;
    acc[3] += x1 * x1; acc[4] += y1 * y1; acc[5] += z1 * z1;
    acc[6] += x2; acc[7] += y2; acc[8] += z2;
    acc[9] += x2 * x2; acc[10] += y2 * y2; acc[11] += z2 * z2;
    acc[12] += DI[n];
  }

  __shared__ float sdata[256];
  __shared__ float res[13];
  for (int k = 0; k < 13; ++k) {
    sdata[tid] = acc[k];
    __syncthreads();
    for (int off = 128; off > 0; off >>= 1) {
      if (tid < off) sdata[tid] += sdata[tid + off];
      __syncthreads();
    }
    if (tid == 0) res[k] = sdata[0];
    __syncthreads();
  }
  if (tid == 0) {
    const float invN = 1.0f / (float)Nn, invD = 1.0f / (float)(Nn - 1);
    float* st = stats + b * 16;
#pragma unroll
    for (int c = 0; c < 3; ++c) {
      const float mu1 = res[c] * invN;
      st[c]     = mu1;
      st[3 + c] = sqrtf(fmaxf(0.f, (res[3 + c] - res[c] * res[c] * invN) * invD));
      const float mu2 = res[6 + c] * invN;
      st[6 + c] = mu2;
      st[9 + c] = sqrtf(fmaxf(0.f, (res[9 + c] - res[6 + c] * res[6 + c] * invN) * invD));
    }
    st[12] = res[12] * invN;  // mean(dist)
  }
}

// ---------------------------------------------------------------------------
// Kernel 3: masked 12x12 Gram matrix via Kronecker structure:
//   M_row contributions = K(n) (x) a a^T,  K built from z2^2, x2*z2, y2*z2,
//   x2^2+y2^2 -> only 4 groups x 10 symmetric a-outer terms = 40 sums.
// ---------------------------------------------------------------------------
__global__ __launch_bounds__(256) void k3_gram(const float* __restrict__ pc1,
                                               const float* __restrict__ pcn,
                                               const float* __restrict__ dist,
                                               const float* __restrict__ stats,
                                               float* __restrict__ gram) {
  const int b = blockIdx.x, tid = threadIdx.x;
  const float* P1 = pc1  + (size_t)b * 4 * Nn;
  const float* PN = pcn  + (size_t)b * 3 * Nn;
  const float* DI = dist + (size_t)b * Nn;
  const float* st = stats + b * 16;
  const float m1x = st[0], m1y = st[1], m1z = st[2];
  const float i1x = 1.f / st[3], i1y = 1.f / st[4], i1z = 1.f / st[5];
  const float m2x = st[6], m2y = st[7], m2z = st[8];
  const float i2x = 1.f / st[9], i2y = 1.f / st[10], i2z = 1.f / st[11];
  const float thr = st[12] * 1.0f + 1e-5f;  // REJECT_RATIO=1, EPS

  float acc[40];
#pragma unroll
  for (int k = 0; k < 40; ++k) acc[k] = 0.f;

  for (int n = tid; n < Nn; n += 256) {
    const float msk = (DI[n] < thr) ? 1.0f : 0.0f;
    float a[4];
    a[0] = (P1[n]          - m1x) * i1x;
    a[1] = (P1[Nn + n]     - m1y) * i1y;
    a[2] = (P1[2 * Nn + n] - m1z) * i1z;
    a[3] = 1.0f;
    const float x2 = (PN[n]          - m2x) * i2x;
    const float y2 = (PN[Nn + n]     - m2y) * i2y;
    const float z2 = (PN[2 * Nn + n] - m2z) * i2z;
    const float czz = msk * z2 * z2;
    const float cxz = msk * x2 * z2;
    const float cyz = msk * y2 * z2;
    const float cw  = msk * (x2 * x2 + y2 * y2);
    int t = 0;
#pragma unroll
    for (int p = 0; p < 4; ++p)
#pragma unroll
      for (int q = p; q < 4; ++q) {
        const float aa = a[p] * a[q];
        acc[t]      += czz * aa;
        acc[10 + t] += cxz * aa;
        acc[20 + t] += cyz * aa;
        acc[30 + t] += cw  * aa;
        ++t;
      }
  }

#pragma unroll
  for (int k = 0; k < 40; ++k)
    for (int off = 1; off <= 16; off <<= 1)
      acc[k] += __shfl_xor(acc[k], off, 32);

  __shared__ float sred[8][40];
  const int wave = tid >> 5, lane = tid & 31;
  if (lane == 0)
    for (int k = 0; k < 40; ++k) sred[wave][k] = acc[k];
  __syncthreads();
  if (tid < 40) {
    float t = 0.f;
    for (int w = 0; w < 8; ++w) t += sred[w][tid];
    gram[b * 40 + tid] = t;
  }
}

// ---------------------------------------------------------------------------
// Kernel 4: per-sample 12x12 Jacobi eigen-solve (min eigenvector == last
// right singular vector of M), transform assembly, rotation fix, quaternion.
// One lane per sample.
// ---------------------------------------------------------------------------
__global__ void k4_solve(const float* __restrict__ stats,
                         const float* __restrict__ gram,
                         float* __restrict__ out) {
  if (threadIdx.x != 0) return;
  const int b = blockIdx.x;
  const float* st = stats + b * 16;
  const float* g  = gram + b * 40;

  auto sidx = [](int p, int q) {
    if (p > q) { int t = p; p = q; q = t; }
    return p * (7 - p) / 2 + q;  // symmetric 4x4 -> 10 packed
  };

  float A[12][12], V[12][12];
  for (int i = 0; i < 12; ++i)
    for (int j = 0; j < 12; ++j) { A[i][j] = 0.f; V[i][j] = (i == j) ? 1.f : 0.f; }

  for (int p = 0; p < 4; ++p)
    for (int q = 0; q < 4; ++q) {
      const int k = sidx(p, q);
      const float szz = g[k], sxz = g[10 + k], syz = g[20 + k], sw = g[30 + k];
      A[p][q]         = szz;    // block (0,0): K00 = z2^2
      A[4 + p][4 + q] = szz;    // block (1,1): K11 = z2^2
      A[p][8 + q]     = -sxz;   // block (0,2): K02 = -x2*z2
      A[8 + q][p]     = -sxz;
      A[4 + p][8 + q] = -syz;   // block (1,2): K12 = -y2*z2
      A[8 + q][4 + p] = -syz;
      A[8 + p][8 + q] = sw;     // block (2,2): K22 = x2^2+y2^2
    }

  // Cyclic Jacobi sweeps.
  for (int sweep = 0; sweep < 16; ++sweep)
    for (int i = 0; i < 11; ++i)
      for (int j = i + 1; j < 12; ++j) {
        const float apq = A[i][j];
        if (fabsf(apq) < 1e-20f) continue;
        const float theta = (A[j][j] - A[i][i]) / (2.0f * apq);
        const float t = ((theta >= 0.f) ? 1.f : -1.f) /
                        (fabsf(theta) + sqrtf(theta * theta + 1.f));
        const float c = 1.0f / sqrtf(t * t + 1.f);
        const float s = t * c;
        for (int k = 0; k < 12; ++k) {
          const float aik = A[i][k], ajk = A[j][k];
          A[i][k] = c * aik - s * ajk;
          A[j][k] = s * aik + c * ajk;
        }
        for (int k = 0; k < 12; ++k) {
          const float aki = A[k][i], akj = A[k][j];
          A[k][i] = c * aki - s * akj;
          A[k][j] = s * aki + c * akj;
        }
        for (int k = 0; k < 12; ++k) {
          const float vki = V[k][i], vkj = V[k][j];
          V[k][i] = c * vki - s * vkj;
          V[k][j] = s * vki + c * vkj;
        }
      }

  int kmin = 0;
  for (int k = 1; k < 12; ++k)
    if (A[k][k] < A[kmin][kmin]) kmin = k;
  float p[12];
  for (int k = 0; k < 12; ++k) p[k] = V[k][kmin];
  if (p[10] < 0.f)
    for (int k = 0; k < 12; ++k) p[k] = -p[k];
  const float nrm = sqrtf(p[8] * p[8] + p[9] * p[9] + p[10] * p[10]);
  for (int k = 0; k < 12; ++k) p[k] /= nrm;

  // T = inv(T2) @ P @ T1
  float P4[4][4] = {{p[0], p[1], p[2], p[3]},
                    {p[4], p[5], p[6], p[7]},
                    {p[8], p[9], p[10], p[11]},
                    {0.f, 0.f, 0.f, 1.f}};
  float T1[4][4] = {{1.f / st[3], 0.f, 0.f, -st[0] / st[3]},
                    {0.f, 1.f / st[4], 0.f, -st[1] / st[4]},
                    {0.f, 0.f, 1.f / st[5], -st[2] / st[5]},
                    {0.f, 0.f, 0.f, 1.f}};
  float I2[4][4] = {{st[9], 0.f, 0.f, st[6]},
                    {0.f, st[10], 0.f, st[7]},
                    {0.f, 0.f, st[11], st[8]},
                    {0.f, 0.f, 0.f, 1.f}};
  float PT[4][4], T[4][4];
  for (int i = 0; i < 4; ++i)
    for (int j = 0; j < 4; ++j) {
      float s = 0.f;
      for (int k = 0; k < 4; ++k) s += P4[i][k] * T1[k][j];
      PT[i][j] = s;
    }
  for (int i = 0; i < 4; ++i)
    for (int j = 0; j < 4; ++j) {
      float s = 0.f;
      for (int k = 0; k < 4; ++k) s += I2[i][k] * PT[k][j];
      T[i][j] = s;
    }

  // Re-orthonormalize R (columns): z = c2/|c2|; x = cross(c1,z)/| |; y = cross(z,x)
  float zx = T[0][2], zy = T[1][2], zz = T[2][2];
  float zn = sqrtf(zx * zx + zy * zy + zz * zz);
  zx /= zn; zy /= zn; zz /= zn;
  float yx = T[0][1], yy = T[1][1], yz = T[2][1];
  float xx = yy * zz - yz * zy;
  float xy = yz * zx - yx * zz;
  float xz = yx * zy - yy * zx;
  const float xn = sqrtf(xx * xx + xy * xy + xz * xz);
  xx /= xn; xy /= xn; xz /= xn;
  yx = zy * xz - zz * xy;
  yy = zz * xx - zx * xz;
  yz = zx * xy - zy * xx;
  T[0][0] = xx; T[1][0] = xy; T[2][0] = xz;
  T[0][1] = yx; T[1][1] = yy; T[2][1] = yz;
  T[0][2] = zx; T[1][2] = zy; T[2][2] = zz;

  // Quaternion
  const float r00 = T[0][0], r11 = T[1][1], r22 = T[2][2];
  const float qw = 0.5f * sqrtf(fmaxf(1e-12f, 1.f + r00 + r11 + r22));
  float qx = 0.5f * sqrtf(fmaxf(0.f, 1.f + r00 - r11 - r22));
  float qy = 0.5f * sqrtf(fmaxf(0.f, 1.f - r00 + r11 - r22));
  float qz = 0.5f * sqrtf(fmaxf(0.f, 1.f - r00 - r11 + r22));
  if (T[2][1] - T[1][2] < 0.f) qx = -qx;
  if (T[0][2] - T[2][0] < 0.f) qy = -qy;
  if (T[1][0] - T[0][1] < 0.f) qz = -qz;

  // Outputs: T (8x4x4) | q (8x4) | t (8x3), flat in return order.
  float* Tout = out + b * 16;
  for (int i = 0; i < 4; ++i)
    for (int j = 0; j < 4; ++j) Tout[i * 4 + j] = T[i][j];
  float* qout = out + Bb * 16 + b * 4;
  qout[0] = qw; qout[1] = qx; qout[2] = qy; qout[3] = qz;
  float* tout = out + Bb * 16 + Bb * 4 + b * 3;
  tout[0] = T[0][3]; tout[1] = T[1][3]; tout[2] = T[2][3];
}

extern "C" void kernel_launch(void* const* d_in, const int* in_sizes, int n_in,
                              void* d_out, int out_size, void* d_ws, size_t ws_size,
                              hipStream_t stream) {
  const float* pc1 = (const float*)d_in[0];
  const float* pc2 = (const float*)d_in[1];
  float* ws    = (float*)d_ws;
  float* pcn   = ws;                              // B*3*N
  float* dist  = pcn + (size_t)Bb * 3 * Nn;       // B*N
  float* stats = dist + (size_t)Bb * Nn;          // B*16
  float* gram  = stats + (size_t)Bb * 16;         // B*40  (~514 KB total)

  k1_flash<<<Bb * (Nn / 64), 128, 0, stream>>>(pc1, pc2, pcn, dist);
  k2_stats<<<Bb, 256, 0, stream>>>(pc1, pcn, dist, stats);
  k3_gram<<<Bb, 256, 0, stream>>>(pc1, pcn, dist, stats, gram);
  k4_solve<<<Bb, 32, 0, stream>>>(stats, gram, (float*)d_out);
}